// MP_block_90374701842948
// MI455X (gfx1250) — compile-verified
//
#include <hip/hip_runtime.h>

typedef float v2f __attribute__((ext_vector_type(2)));
typedef float v8f __attribute__((ext_vector_type(8)));

#define HD  64     // hidden dim
#define LP  68     // padded LDS row stride for activation staging
#define WPR 160    // dwords per weight row-pair (128 data + 32 pad -> bank shift 32)

__device__ __forceinline__ void wmma4(const v2f& a, const v2f& b, v8f& c) {
  c = __builtin_amdgcn_wmma_f32_16x16x4_f32(false, a, false, b, (short)0, c,
                                            false, false);
}

// B-fragment: rows [kk, kk+1] at column col, from pair-interleaved LDS weights.
// Pair p = kk/2 holds {W[2p][c], W[2p+1][c]} at w + p*WPR + 2c  -> one b64 load.
__device__ __forceinline__ v2f bfrag(const float* __restrict__ w, int kk,
                                     int col) {
  return *(const v2f*)(w + (kk >> 1) * WPR + 2 * col);
}

__global__ void zero_kernel(float* __restrict__ p, long long n) {
  long long i = (long long)blockIdx.x * blockDim.x + threadIdx.x;
  long long stride = (long long)gridDim.x * blockDim.x;
  for (; i < n; i += stride) p[i] = 0.0f;
}

__global__ void count_kernel(const int* __restrict__ eidx,
                             float* __restrict__ cnt, int E) {
  long long i = (long long)blockIdx.x * blockDim.x + threadIdx.x;
  long long stride = (long long)gridDim.x * blockDim.x;
  for (; i < E; i += stride)
    atomicAdd(&cnt[eidx[E + i]], 1.0f);
}

// Fused edge pipeline: edge_mlp (K=192 -> relu -> K=64) -> new edge_attr,
// then node_mlp_1 (K=128 -> relu -> K=64) -> messages scattered into agg.
__global__ void __launch_bounds__(256)
edge_kernel(const float* __restrict__ x, const int* __restrict__ eidx,
            const float* __restrict__ ea_in, float* __restrict__ edge_out,
            float* __restrict__ agg,
            const float* __restrict__ eW1, const float* __restrict__ eb1,
            const float* __restrict__ eW2, const float* __restrict__ eb2,
            const float* __restrict__ nW1, const float* __restrict__ nb1,
            const float* __restrict__ nW2, const float* __restrict__ nb2,
            int E) {
  extern __shared__ float sm[];
  float* wghts = sm;                        // 224 row-pairs * WPR
  float* bias  = sm + 224 * WPR;            // 4 * 64
  float* stag  = bias + 4 * 64;             // 8 waves * 2 * 16 * LP
  int*   cids  = (int*)(stag + 8 * 2 * 16 * LP);  // 8 waves * 16

  const int tid = threadIdx.x;
  // Pair-interleaved weight stage: wghts[(r/2)*WPR + 2c + (r&1)] = W[r][c].
  for (int i = tid; i < 448 * 64; i += 256) {
    int r = i >> 6, c = i & 63;
    float v;
    if (r < 192)      v = eW1[i];
    else if (r < 256) v = eW2[i - 192 * 64];
    else if (r < 384) v = nW1[i - 256 * 64];
    else              v = nW2[i - 384 * 64];
    wghts[(r >> 1) * WPR + 2 * c + (r & 1)] = v;
  }
  if (tid < 64) {
    bias[tid]       = eb1[tid];
    bias[64 + tid]  = eb2[tid];
    bias[128 + tid] = nb1[tid];
    bias[192 + tid] = nb2[tid];
  }
  __syncthreads();

  const int lane = tid & 31;
  const int wave = tid >> 5;
  const int l15  = lane & 15;
  const int half = lane >> 4;

  float* hbuf = stag + wave * (2 * 16 * LP);
  float* ebuf = hbuf + 16 * LP;
  int*   myc  = cids + wave * 16;

  const float* wE1 = wghts;                 // rows   0..191 -> pairs   0..95
  const float* wE2 = wghts + 96 * WPR;      // rows 192..255 -> pairs  96..127
  const float* wN1 = wghts + 128 * WPR;     // rows 256..383 -> pairs 128..191
  const float* wN2 = wghts + 192 * WPR;     // rows 384..447 -> pairs 192..223

  float bE1r[4], bE2r[4], bN1r[4], bN2r[4];
  #pragma unroll
  for (int t = 0; t < 4; ++t) {
    bE1r[t] = bias[t * 16 + l15];
    bE2r[t] = bias[64 + t * 16 + l15];
    bN1r[t] = bias[128 + t * 16 + l15];
    bN2r[t] = bias[192 + t * 16 + l15];
  }

  const v8f vz = {0, 0, 0, 0, 0, 0, 0, 0};
  const int numTiles = (E + 15) >> 4;

  for (int tile = blockIdx.x * 8 + wave; tile < numTiles;
       tile += gridDim.x * 8) {
    const int base = tile << 4;
    const int er   = min(base + l15, E - 1);
    const int rsrc = eidx[er];
    const int csrc = eidx[E + er];
    if (lane < 16) myc[l15] = csrc;
    const float* pxr = x + (size_t)rsrc * HD + 2 * half;
    const float* pxc = x + (size_t)csrc * HD + 2 * half;
    const float* pea = ea_in + (size_t)er * HD + 2 * half;

    // Preload x[row] A-fragments once; reused by GEMM1 and GEMM3.
    v2f xr[16];
    #pragma unroll
    for (int j = 0; j < 16; ++j)
      xr[j] = *(const v2f*)(pxr + j * 4);

    v8f acc[4];
    // ---- GEMM1: [16x192] @ eW1 (three 64-wide segments) ----
    #pragma unroll
    for (int t = 0; t < 4; ++t) acc[t] = vz;
    #pragma unroll                         // full: keeps xr[] statically indexed
    for (int j = 0; j < 16; ++j) {
      const int kk = j * 4 + 2 * half;
      #pragma unroll
      for (int t = 0; t < 4; ++t)
        wmma4(xr[j], bfrag(wE1, kk, t * 16 + l15), acc[t]);
    }
    #pragma unroll 4
    for (int j = 0; j < 16; ++j) {
      const int kk = 64 + j * 4 + 2 * half;
      v2f a = *(const v2f*)(pxc + j * 4);
      #pragma unroll
      for (int t = 0; t < 4; ++t)
        wmma4(a, bfrag(wE1, kk, t * 16 + l15), acc[t]);
    }
    #pragma unroll 4
    for (int j = 0; j < 16; ++j) {
      const int kk = 128 + j * 4 + 2 * half;
      v2f a = __builtin_nontemporal_load((const v2f*)(pea + j * 4));
      #pragma unroll
      for (int t = 0; t < 4; ++t)
        wmma4(a, bfrag(wE1, kk, t * 16 + l15), acc[t]);
    }
    #pragma unroll
    for (int r = 0; r < 8; ++r) {
      float* hrow = hbuf + (r + 8 * half) * LP + l15;
      #pragma unroll
      for (int t = 0; t < 4; ++t)
        hrow[t * 16] = fmaxf(acc[t][r] + bE1r[t], 0.0f);
    }

    // ---- GEMM2: [16x64] @ eW2 -> new edge_attr ----
    #pragma unroll
    for (int t = 0; t < 4; ++t) acc[t] = vz;
    const float* ha = hbuf + l15 * LP + 2 * half;
    #pragma unroll 4
    for (int j = 0; j < 16; ++j) {
      const int kk = j * 4 + 2 * half;
      v2f a = *(const v2f*)(ha + j * 4);
      #pragma unroll
      for (int t = 0; t < 4; ++t)
        wmma4(a, bfrag(wE2, kk, t * 16 + l15), acc[t]);
    }
    #pragma unroll
    for (int r = 0; r < 8; ++r) {
      const int m = r + 8 * half;
      float* erow = ebuf + m * LP + l15;
      float* gout = edge_out + (size_t)(base + m) * HD + l15;
      const bool ok = (base + m) < E;
      #pragma unroll
      for (int t = 0; t < 4; ++t) {
        float v = acc[t][r] + bE2r[t];
        erow[t * 16] = v;
        if (ok) __builtin_nontemporal_store(v, gout + t * 16);
      }
    }

    // ---- GEMM3: [16x128] (x[row] ++ new_edge) @ n1W1 ----
    #pragma unroll
    for (int t = 0; t < 4; ++t) acc[t] = vz;
    #pragma unroll                         // full: keeps xr[] statically indexed
    for (int j = 0; j < 16; ++j) {
      const int kk = j * 4 + 2 * half;
      #pragma unroll
      for (int t = 0; t < 4; ++t)
        wmma4(xr[j], bfrag(wN1, kk, t * 16 + l15), acc[t]);
    }
    const float* ea2 = ebuf + l15 * LP + 2 * half;
    #pragma unroll 4
    for (int j = 0; j < 16; ++j) {
      const int kk = 64 + j * 4 + 2 * half;
      v2f a = *(const v2f*)(ea2 + j * 4);
      #pragma unroll
      for (int t = 0; t < 4; ++t)
        wmma4(a, bfrag(wN1, kk, t * 16 + l15), acc[t]);
    }
    #pragma unroll
    for (int r = 0; r < 8; ++r) {
      float* hrow = hbuf + (r + 8 * half) * LP + l15;
      #pragma unroll
      for (int t = 0; t < 4; ++t)
        hrow[t * 16] = fmaxf(acc[t][r] + bN1r[t], 0.0f);
    }

    // ---- GEMM4: [16x64] @ n1W2 -> message, scatter-add ----
    #pragma unroll
    for (int t = 0; t < 4; ++t) acc[t] = vz;
    #pragma unroll 4
    for (int j = 0; j < 16; ++j) {
      const int kk = j * 4 + 2 * half;
      v2f a = *(const v2f*)(ha + j * 4);
      #pragma unroll
      for (int t = 0; t < 4; ++t)
        wmma4(a, bfrag(wN2, kk, t * 16 + l15), acc[t]);
    }
    #pragma unroll
    for (int r = 0; r < 8; ++r) {
      const int m = r + 8 * half;
      if (base + m < E) {
        float* arow = agg + (size_t)myc[m] * HD + l15;
        #pragma unroll
        for (int t = 0; t < 4; ++t)
          atomicAdd(arow + t * 16, acc[t][r] + bN2r[t]);
      }
    }
  }
}

// Node update: x_out = mlp2(concat(x, agg/max(cnt,1)))
__global__ void __launch_bounds__(256)
node_kernel(const float* __restrict__ x_in, const float* __restrict__ agg,
            const float* __restrict__ cnt, float* __restrict__ x_out,
            const float* __restrict__ W1, const float* __restrict__ b1,
            const float* __restrict__ W2, const float* __restrict__ b2,
            int N) {
  extern __shared__ float sm[];
  float* wghts = sm;                 // 96 row-pairs * WPR
  float* bias  = sm + 96 * WPR;      // 2 * 64
  float* stag  = bias + 128;         // 8 waves * 16 * LP

  const int tid = threadIdx.x;
  for (int i = tid; i < 192 * 64; i += 256) {
    int r = i >> 6, c = i & 63;
    float v = (r < 128) ? W1[i] : W2[i - 128 * 64];
    wghts[(r >> 1) * WPR + 2 * c + (r & 1)] = v;
  }
  if (tid < 64) { bias[tid] = b1[tid]; bias[64 + tid] = b2[tid]; }
  __syncthreads();

  const int lane = tid & 31;
  const int wave = tid >> 5;
  const int l15  = lane & 15;
  const int half = lane >> 4;
  float* hbuf = stag + wave * (16 * LP);
  const float* wA = wghts;             // rows 0..127 -> pairs 0..63
  const float* wB = wghts + 64 * WPR;  // rows 128..191 -> pairs 64..95

  float b1r[4], b2r[4];
  #pragma unroll
  for (int t = 0; t < 4; ++t) {
    b1r[t] = bias[t * 16 + l15];
    b2r[t] = bias[64 + t * 16 + l15];
  }

  const v8f vz = {0, 0, 0, 0, 0, 0, 0, 0};
  const int numTiles = (N + 15) >> 4;

  for (int tile = blockIdx.x * 8 + wave; tile < numTiles;
       tile += gridDim.x * 8) {
    const int base = tile << 4;
    const int nd = min(base + l15, N - 1);
    const float* px = x_in + (size_t)nd * HD + 2 * half;
    const float* pa = agg + (size_t)nd * HD + 2 * half;
    const float invd = 1.0f / fmaxf(cnt[nd], 1.0f);

    v8f acc[4];
    #pragma unroll
    for (int t = 0; t < 4; ++t) acc[t] = vz;
    #pragma unroll 4
    for (int j = 0; j < 16; ++j) {
      const int kk = j * 4 + 2 * half;
      v2f a = *(const v2f*)(px + j * 4);
      #pragma unroll
      for (int t = 0; t < 4; ++t)
        wmma4(a, bfrag(wA, kk, t * 16 + l15), acc[t]);
    }
    #pragma unroll 4
    for (int j = 0; j < 16; ++j) {
      const int kk = 64 + j * 4 + 2 * half;
      v2f a = *(const v2f*)(pa + j * 4);
      a *= invd;
      #pragma unroll
      for (int t = 0; t < 4; ++t)
        wmma4(a, bfrag(wA, kk, t * 16 + l15), acc[t]);
    }
    #pragma unroll
    for (int r = 0; r < 8; ++r) {
      float* hrow = hbuf + (r + 8 * half) * LP + l15;
      #pragma unroll
      for (int t = 0; t < 4; ++t)
        hrow[t * 16] = fmaxf(acc[t][r] + b1r[t], 0.0f);
    }

    #pragma unroll
    for (int t = 0; t < 4; ++t) acc[t] = vz;
    const float* ha = hbuf + l15 * LP + 2 * half;
    #pragma unroll 4
    for (int j = 0; j < 16; ++j) {
      const int kk = j * 4 + 2 * half;
      v2f a = *(const v2f*)(ha + j * 4);
      #pragma unroll
      for (int t = 0; t < 4; ++t)
        wmma4(a, bfrag(wB, kk, t * 16 + l15), acc[t]);
    }
    #pragma unroll
    for (int r = 0; r < 8; ++r) {
      const int m = r + 8 * half;
      if (base + m < N) {
        float* orow = x_out + (size_t)(base + m) * HD + l15;
        #pragma unroll
        for (int t = 0; t < 4; ++t)
          orow[t * 16] = acc[t][r] + b2r[t];
      }
    }
  }
}

extern "C" void kernel_launch(void* const* d_in, const int* in_sizes, int n_in,
                              void* d_out, int out_size, void* d_ws,
                              size_t ws_size, hipStream_t stream) {
  const float* x0   = (const float*)d_in[0];
  const int*   eidx = (const int*)d_in[1];
  const float* ea0  = (const float*)d_in[2];
  const float* eW1  = (const float*)d_in[3];
  const float* eb1  = (const float*)d_in[4];
  const float* eW2  = (const float*)d_in[5];
  const float* eb2  = (const float*)d_in[6];
  const float* n1W1 = (const float*)d_in[7];
  const float* n1b1 = (const float*)d_in[8];
  const float* n1W2 = (const float*)d_in[9];
  const float* n1b2 = (const float*)d_in[10];
  const float* n2W1 = (const float*)d_in[11];
  const float* n2b1 = (const float*)d_in[12];
  const float* n2W2 = (const float*)d_in[13];
  const float* n2b2 = (const float*)d_in[14];

  const int N = in_sizes[0] / HD;
  const int E = in_sizes[2] / HD;

  float* out_x = (float*)d_out;
  float* out_e = out_x + (size_t)N * HD;

  float* edge_tmp = (float*)d_ws;
  float* x_tmp    = edge_tmp + (size_t)E * HD;
  float* agg      = x_tmp + (size_t)N * HD;
  float* cnt      = agg + (size_t)N * HD;

  const size_t edgeLds = (size_t)(224 * WPR + 4 * 64 + 8 * 2 * 16 * LP) * 4 +
                         8 * 16 * 4;
  const size_t nodeLds = (size_t)(96 * WPR + 2 * 64 + 8 * 16 * LP) * 4;

  zero_kernel<<<256, 256, 0, stream>>>(cnt, (long long)N);
  count_kernel<<<(E + 255) / 256, 256, 0, stream>>>(eidx, cnt, E);

  const int tilesE = (E + 15) / 16;
  const int tilesN = (N + 15) / 16;
  int gridE = (tilesE + 7) / 8; if (gridE > 1536) gridE = 1536;
  int gridN = (tilesN + 7) / 8; if (gridN > 1024) gridN = 1024;

  for (int l = 0; l < 2; ++l) {
    const float* xin  = (l == 0) ? x0 : x_tmp;
    const float* ein  = (l == 0) ? ea0 : edge_tmp;
    float*       eout = (l == 0) ? edge_tmp : out_e;
    float*       xout = (l == 0) ? x_tmp : out_x;

    zero_kernel<<<512, 256, 0, stream>>>(agg, (long long)N * HD);
    edge_kernel<<<gridE, 256, edgeLds, stream>>>(
        xin, eidx, ein, eout, agg,
        eW1 + (size_t)l * 192 * 64, eb1 + (size_t)l * 64,
        eW2 + (size_t)l * 64 * 64,  eb2 + (size_t)l * 64,
        n1W1 + (size_t)l * 128 * 64, n1b1 + (size_t)l * 64,
        n1W2 + (size_t)l * 64 * 64,  n1b2 + (size_t)l * 64, E);
    node_kernel<<<gridN, 256, nodeLds, stream>>>(
        xin, agg, cnt, xout,
        n2W1 + (size_t)l * 128 * 64, n2b1 + (size_t)l * 64,
        n2W2 + (size_t)l * 64 * 64,  n2b2 + (size_t)l * 64, N);
  }
}